// GNN_28217935135024
// MI455X (gfx1250) — compile-verified
//
#include <hip/hip_runtime.h>

#define N_NODES 50000
#define N_EDGES 800000
#define DIM     128
#define HID     128
#define HALF    64
#define NOUT    10
#define NLAYERS 3
#define EPSLN   1e-5f

typedef __attribute__((ext_vector_type(2))) float v2f;
typedef __attribute__((ext_vector_type(8))) float v8f;

// ---------------------------------------------------------------- utilities
__global__ void fill_zero_kernel(float* __restrict__ p, int n) {
  int i = blockIdx.x * blockDim.x + threadIdx.x;
  if (i < n) p[i] = 0.0f;
}

// deg[dst] += 1 for every edge (float so we can reuse buffer for 1/deg)
__global__ void degree_kernel(const int* __restrict__ ei, float* __restrict__ deg) {
  int e = blockIdx.x * blockDim.x + threadIdx.x;
  if (e < N_EDGES) atomicAdd(&deg[ei[N_EDGES + e]], 1.0f);
}

__global__ void invdeg_kernel(float* __restrict__ deg) {
  int i = blockIdx.x * blockDim.x + threadIdx.x;
  if (i < N_NODES) deg[i] = 1.0f / fmaxf(deg[i], 1.0f);
}

// ------------------------------------------------- edge scatter aggregation
// One wave (32 lanes) per edge; each lane handles 4 contiguous floats so a
// wave covers one 512B feature row (coalesced gather, L2-resident atomics).
__global__ void aggregate_kernel(const int* __restrict__ ei,
                                 const float* __restrict__ h,
                                 float* __restrict__ agg) {
  long t = (long)blockIdx.x * blockDim.x + threadIdx.x;
  long e = t >> 5;
  int lane = (int)(t & 31);
  if (e >= N_EDGES) return;
  int src = ei[e];
  int dst = ei[N_EDGES + e];
  const float4 v = *(const float4*)(h + (long)src * HID + lane * 4);
  float* d = agg + (long)dst * HID + lane * 4;
  atomicAdd(d + 0, v.x);
  atomicAdd(d + 1, v.y);
  atomicAdd(d + 2, v.z);
  atomicAdd(d + 3, v.w);
}

// -------------------------------- fused SAGEConv + LayerNorm + residual/ReLU
// One wave per 16-row tile. acc = (agg*invdeg)@Wl + h@Wr (WMMA f32 16x16x4,
// 512 wmma/block), then bias + in-register LayerNorm (half-wave shfl_xor
// butterflies) + residual + ReLU.
__global__ __launch_bounds__(32) void sage_layer_kernel(
    const float* __restrict__ h, const float* __restrict__ agg,
    const float* __restrict__ invdeg,
    const float* __restrict__ Wl, const float* __restrict__ bl,
    const float* __restrict__ Wr,
    const float* __restrict__ gamma, const float* __restrict__ beta,
    float* __restrict__ hout) {
  const int lane = threadIdx.x;
  const int m  = lane & 15;   // row (A) / col (B,C) within tile
  const int hi = lane >> 4;   // half-wave: selects K pair (A/B), row+8 (C/D)
  const int row0 = blockIdx.x * 16;
  const int rA = row0 + m;
  const float inv = invdeg[rA];

  v8f acc[8];
#pragma unroll
  for (int j = 0; j < 8; ++j) acc[j] = (v8f)(0.0f);

  const float* hrow = h + (long)rA * HID;
  const float* arow = agg + (long)rA * HID;

#pragma unroll 2
  for (int k0 = 0; k0 < HID; k0 += 4) {
    const int ka = k0 + 2 * hi;
    // neighbor path: A = agg/deg
    v2f a;
    a.x = arow[ka] * inv;
    a.y = arow[ka + 1] * inv;
#pragma unroll
    for (int j = 0; j < 8; ++j) {
      v2f b;
      b.x = Wl[ka * HID + j * 16 + m];
      b.y = Wl[(ka + 1) * HID + j * 16 + m];
      acc[j] = __builtin_amdgcn_wmma_f32_16x16x4_f32(
          false, a, false, b, (short)0, acc[j], false, false);
    }
    // root path: A = h
    a.x = hrow[ka];
    a.y = hrow[ka + 1];
#pragma unroll
    for (int j = 0; j < 8; ++j) {
      v2f b;
      b.x = Wr[ka * HID + j * 16 + m];
      b.y = Wr[(ka + 1) * HID + j * 16 + m];
      acc[j] = __builtin_amdgcn_wmma_f32_16x16x4_f32(
          false, a, false, b, (short)0, acc[j], false, false);
    }
  }

  // bias bl[n], n = j*16+m (same for all 8 rows held in the accumulator)
#pragma unroll
  for (int j = 0; j < 8; ++j) {
    const float bj = bl[j * 16 + m];
#pragma unroll
    for (int i = 0; i < 8; ++i) acc[j][i] += bj;
  }

  // LayerNorm stats for row r = row0 + i + 8*hi; reduce 8 local cols then
  // 16 lanes of this half-wave (xor masks 1/2/4/8 stay inside the half).
  float mu[8], rstd[8];
#pragma unroll
  for (int i = 0; i < 8; ++i) {
    float s = 0.0f, ss = 0.0f;
#pragma unroll
    for (int j = 0; j < 8; ++j) {
      const float v = acc[j][i];
      s += v;
      ss += v * v;
    }
#pragma unroll
    for (int mask = 8; mask >= 1; mask >>= 1) {
      s  += __shfl_xor(s,  mask, 32);
      ss += __shfl_xor(ss, mask, 32);
    }
    const float mean = s * (1.0f / HID);
    const float var  = ss * (1.0f / HID) - mean * mean;
    mu[i] = mean;
    rstd[i] = rsqrtf(var + EPSLN);
  }

  // affine + residual + ReLU, store
#pragma unroll
  for (int j = 0; j < 8; ++j) {
    const int n = j * 16 + m;
    const float g = gamma[n];
    const float bb = beta[n];
#pragma unroll
    for (int i = 0; i < 8; ++i) {
      const int r = row0 + i + 8 * hi;
      float y = g * (acc[j][i] - mu[i]) * rstd[i] + bb + h[(long)r * HID + n];
      hout[(long)r * HID + n] = fmaxf(y, 0.0f);
    }
  }
}

// ----------------------------------- MLP head: relu(h@W1+b1)@W2+b2 -> [N,10]
__global__ __launch_bounds__(32) void mlp_kernel(
    const float* __restrict__ h,
    const float* __restrict__ W1, const float* __restrict__ b1,
    const float* __restrict__ W2, const float* __restrict__ b2,
    float* __restrict__ out) {
  __shared__ float mid[16 * HALF];
  const int lane = threadIdx.x;
  const int m  = lane & 15;
  const int hi = lane >> 4;
  const int row0 = blockIdx.x * 16;

  v8f acc[4];
#pragma unroll
  for (int j = 0; j < 4; ++j) acc[j] = (v8f)(0.0f);

  const float* hrow = h + (long)(row0 + m) * HID;
#pragma unroll 2
  for (int k0 = 0; k0 < HID; k0 += 4) {
    const int ka = k0 + 2 * hi;
    v2f a;
    a.x = hrow[ka];
    a.y = hrow[ka + 1];
#pragma unroll
    for (int j = 0; j < 4; ++j) {
      v2f b;
      b.x = W1[ka * HALF + j * 16 + m];
      b.y = W1[(ka + 1) * HALF + j * 16 + m];
      acc[j] = __builtin_amdgcn_wmma_f32_16x16x4_f32(
          false, a, false, b, (short)0, acc[j], false, false);
    }
  }

  // bias + ReLU, stage to LDS in row-major [16][64]
#pragma unroll
  for (int j = 0; j < 4; ++j) {
    const float bj = b1[j * 16 + m];
#pragma unroll
    for (int i = 0; i < 8; ++i)
      mid[(i + 8 * hi) * HALF + j * 16 + m] = fmaxf(acc[j][i] + bj, 0.0f);
  }
  __syncthreads();

  // tiny [16x64]@[64x10] GEMM in VALU: 160 outputs over 32 lanes
  for (int t = lane; t < 16 * NOUT; t += 32) {
    const int r = t / NOUT;
    const int o = t - r * NOUT;
    float s = b2[o];
#pragma unroll 8
    for (int k = 0; k < HALF; ++k) s += mid[r * HALF + k] * W2[k * NOUT + o];
    out[(long)(row0 + r) * NOUT + o] = s;
  }
}

// ---------------------------------------------------------------- dispatch
extern "C" void kernel_launch(void* const* d_in, const int* in_sizes, int n_in,
                              void* d_out, int out_size, void* d_ws, size_t ws_size,
                              hipStream_t stream) {
  const float* x     = (const float*)d_in[0];
  const int*   ei    = (const int*)d_in[1];
  const float* Wl    = (const float*)d_in[2];
  const float* bl    = (const float*)d_in[3];
  const float* Wr    = (const float*)d_in[4];
  const float* gamma = (const float*)d_in[5];
  const float* beta  = (const float*)d_in[6];
  const float* W1    = (const float*)d_in[7];
  const float* b1    = (const float*)d_in[8];
  const float* W2    = (const float*)d_in[9];
  const float* b2    = (const float*)d_in[10];
  float* out = (float*)d_out;

  float* ws  = (float*)d_ws;
  float* deg = ws;                                  // N floats (-> 1/deg)
  float* hA  = deg + N_NODES;                       // N*H
  float* hB  = hA + (size_t)N_NODES * HID;          // N*H
  float* agg = hB + (size_t)N_NODES * HID;          // N*H

  const int NT = N_NODES / 16;                      // 3125, exact
  const int featN = N_NODES * HID;

  // in-degree -> 1/max(deg,1)
  fill_zero_kernel<<<(N_NODES + 255) / 256, 256, 0, stream>>>(deg, N_NODES);
  degree_kernel<<<(N_EDGES + 255) / 256, 256, 0, stream>>>(ei, deg);
  invdeg_kernel<<<(N_NODES + 255) / 256, 256, 0, stream>>>(deg);

  const long aggThreads = (long)N_EDGES * 32;
  const int aggBlocks = (int)((aggThreads + 255) / 256);

  const float* hin = x;
  float* bufs[NLAYERS] = {hA, hB, hA};
  for (int l = 0; l < NLAYERS; ++l) {
    fill_zero_kernel<<<(featN + 255) / 256, 256, 0, stream>>>(agg, featN);
    aggregate_kernel<<<aggBlocks, 256, 0, stream>>>(ei, hin, agg);
    sage_layer_kernel<<<NT, 32, 0, stream>>>(
        hin, agg, deg,
        Wl + (size_t)l * DIM * HID, bl + (size_t)l * HID,
        Wr + (size_t)l * DIM * HID,
        gamma + (size_t)l * HID, beta + (size_t)l * HID,
        bufs[l]);
    hin = bufs[l];
  }
  mlp_kernel<<<NT, 32, 0, stream>>>(hin, W1, b1, W2, b2, out);
}